// GaussianSplattingCompliance_61065845014565
// MI455X (gfx1250) — compile-verified
//
#include <hip/hip_runtime.h>

typedef __attribute__((ext_vector_type(2))) float v2f;
typedef __attribute__((ext_vector_type(8))) float v8f;

#define NG          256
#define NEL         262144
#define NTILES      (NEL / 16)
#define THREADS     256
#define MM_BLOCKS   256
#define MAIN_BLOCKS 1024

// workspace layout (bytes):
//   [0,16)   : 4 encoded-uint atomics: min_x, min_y, max_x, max_y
//   [64, 64+NG*8*4)          : per-gaussian quadratic-form coeffs (256 x 8 f32)
//   [WS_PART_OFF, +12*MAIN_BLOCKS) : per-block partial sums (comp, H, overlap)
#define WS_COEF_OFF 64
#define WS_PART_OFF (WS_COEF_OFF + NG * 8 * 4)

#define EMIN_F 1e-9f

// ---- order-preserving float<->uint encoding (works for negatives too) ----
__device__ __forceinline__ unsigned enc_f(float f) {
  unsigned u = __float_as_uint(f);
  return (u & 0x80000000u) ? ~u : (u | 0x80000000u);
}
__device__ __forceinline__ float dec_f(unsigned u) {
  unsigned v = (u & 0x80000000u) ? (u & 0x7FFFFFFFu) : ~u;
  return __uint_as_float(v);
}

// fast 1/(1+exp(t)) : v_mul + v_exp_f32 + v_add + v_rcp_f32 (no IEEE div fixups)
__device__ __forceinline__ float sig_of_neg(float t) {
  return __builtin_amdgcn_rcpf(1.0f + __expf(t));
}

__device__ __forceinline__ void block_reduce3(float& a, float& b, float& c) {
  for (int off = 16; off; off >>= 1) {
    a += __shfl_xor(a, off);
    b += __shfl_xor(b, off);
    c += __shfl_xor(c, off);
  }
  __shared__ float sa[8], sb[8], sc[8];
  const int lane = threadIdx.x & 31;
  const int w = threadIdx.x >> 5;
  if (lane == 0) { sa[w] = a; sb[w] = b; sc[w] = c; }
  __syncthreads();
  if (threadIdx.x == 0) {
    const int nw = blockDim.x >> 5;
    for (int i = 1; i < nw; ++i) { a += sa[i]; b += sb[i]; c += sc[i]; }
  }
}

// ---------------- kernel 1: init min/max atomics ----------------
__global__ void k_init(unsigned* hdr) {
  if (threadIdx.x == 0) {
    hdr[0] = 0xFFFFFFFFu;  // min_x (encoded)
    hdr[1] = 0xFFFFFFFFu;  // min_y
    hdr[2] = 0u;           // max_x
    hdr[3] = 0u;           // max_y
  }
}

// ---------------- kernel 2: coordinate min/max ----------------
__global__ void __launch_bounds__(THREADS) k_minmax(const float* __restrict__ coords,
                                                    unsigned* hdr) {
  float mnx = INFINITY, mny = INFINITY, mxx = -INFINITY, mxy = -INFINITY;
  for (int e = blockIdx.x * blockDim.x + threadIdx.x; e < NEL;
       e += gridDim.x * blockDim.x) {
    const float x = coords[2 * e], y = coords[2 * e + 1];
    mnx = fminf(mnx, x); mny = fminf(mny, y);
    mxx = fmaxf(mxx, x); mxy = fmaxf(mxy, y);
  }
  for (int off = 16; off; off >>= 1) {
    mnx = fminf(mnx, __shfl_xor(mnx, off));
    mny = fminf(mny, __shfl_xor(mny, off));
    mxx = fmaxf(mxx, __shfl_xor(mxx, off));
    mxy = fmaxf(mxy, __shfl_xor(mxy, off));
  }
  __shared__ float s0[8], s1[8], s2[8], s3[8];
  const int lane = threadIdx.x & 31, w = threadIdx.x >> 5;
  if (lane == 0) { s0[w] = mnx; s1[w] = mny; s2[w] = mxx; s3[w] = mxy; }
  __syncthreads();
  if (threadIdx.x == 0) {
    const int nw = blockDim.x >> 5;
    for (int i = 1; i < nw; ++i) {
      mnx = fminf(mnx, s0[i]); mny = fminf(mny, s1[i]);
      mxx = fmaxf(mxx, s2[i]); mxy = fmaxf(mxy, s3[i]);
    }
    atomicMin(&hdr[0], enc_f(mnx));
    atomicMin(&hdr[1], enc_f(mny));
    atomicMax(&hdr[2], enc_f(mxx));
    atomicMax(&hdr[3], enc_f(mxy));
  }
}

// ---------------- kernel 3: per-gaussian quadratic-form coefficients ----------
// z^2[n,e] = c0*x^2 + c1*y^2 + c2*x*y + c3*x + c4*y + c5  (c6=c7=0 padding)
__global__ void k_coeffs(const float* __restrict__ Wscale,
                         const float* __restrict__ Wshape,
                         const float* __restrict__ Wrot,
                         const float* __restrict__ Woff,
                         const unsigned* __restrict__ hdr,
                         float* __restrict__ C) {
  const int n = threadIdx.x;
  if (n >= NG) return;
  const float cminx = dec_f(hdr[0]), cminy = dec_f(hdr[1]);
  const float cmaxx = dec_f(hdr[2]), cmaxy = dec_f(hdr[3]);

  const float sig_s  = 1.f / (1.f + __expf(-Wscale[n]));
  const float bscale = (0.06f - 0.027f) * sig_s + 0.027f;
  const float ratio  = 2.0f / (1.f + __expf(-Wshape[n])) + 0.5f;
  const float rot    = -1.57079632679f + 3.14159265359f / (1.f + __expf(-Wrot[n]));
  const float ox = cminx + (cmaxx - cminx) / (1.f + __expf(-Woff[2 * n]));
  const float oy = cminy + (cmaxy - cminy) / (1.f + __expf(-Woff[2 * n + 1]));

  const float a = 1.f / (1.f + 1e-6f);               // inv_cov_diag[0]
  const float b = 1.f / (ratio * ratio + 1e-6f);     // inv_cov_diag[1]
  const float s = bscale + 1e-8f;
  const float inv_s2 = 1.f / (s * s);
  const float cr = cosf(rot), sr = sinf(rot);

  const float P  = (a * cr * cr + b * sr * sr) * inv_s2;
  const float Q  = (a * sr * sr + b * cr * cr) * inv_s2;
  const float Rc = 2.f * cr * sr * (b - a) * inv_s2;

  C[n * 8 + 0] = P;
  C[n * 8 + 1] = Q;
  C[n * 8 + 2] = Rc;
  C[n * 8 + 3] = -(2.f * P * ox + Rc * oy);
  C[n * 8 + 4] = -(2.f * Q * oy + Rc * ox);
  C[n * 8 + 5] = P * ox * ox + Q * oy * oy + Rc * ox * oy;
  C[n * 8 + 6] = 0.f;
  C[n * 8 + 7] = 0.f;
}

// ---------------- kernel 4: main WMMA pairwise kernel ----------------
// Each wave owns one 16-element B tile; loops over 16 gaussian A tiles.
// z2 tile (16 gaussians x 16 elements) via 2x V_WMMA_F32_16X16X4_F32,
// then elementwise fast sqrt/sigmoids, reduce gaussian axis in-lane +
// shfl_xor(16) across lane halves (C/D layout: lanes 16-31 hold rows 8-15).
__global__ void __launch_bounds__(THREADS) k_main(const float* __restrict__ coords,
                                                  const float* __restrict__ ce,
                                                  const float* __restrict__ coef,
                                                  float* __restrict__ part) {
  __shared__ float ldsC[NG * 8];
  for (int i = threadIdx.x; i < NG * 8; i += blockDim.x) ldsC[i] = coef[i];
  __syncthreads();

  const int lane = threadIdx.x & 31;
  const int ecol = lane & 15;        // element column within tile (also A row)
  const int half = lane >> 4;        // 0: K0/K1, 1: K2/K3
  const int wavesPerBlock = THREADS >> 5;
  const int gwave = blockIdx.x * wavesPerBlock + (threadIdx.x >> 5);
  const int nwaves = gridDim.x * wavesPerBlock;
  const float mask = (half == 0) ? 1.0f : 0.0f;

  float acc_comp = 0.f, acc_H = 0.f, acc_ov = 0.f;

  for (int t = gwave; t < NTILES; t += nwaves) {
    const int e = t * 16 + ecol;
    const float x = coords[2 * e];
    const float y = coords[2 * e + 1];
    const float cee = ce[e];

    // B = features^T (K x 16). Branch-free selects keep EXEC all-ones.
    v2f B0, B1;
    B0.x = half ? (x * y) : (x * x);   // K2 | K0
    B0.y = half ? x       : (y * y);   // K3 | K1
    B1.x = half ? 0.f     : y;         // K6 | K4
    B1.y = half ? 0.f     : 1.f;       // K7 | K5

    float ks = 0.f, hs = 0.f;
#pragma unroll 2
    for (int g = 0; g < 16; ++g) {
      const int G = g * 16 + ecol;
      v2f A0, A1;
      A0.x = ldsC[G * 8 + 2 * half];
      A0.y = ldsC[G * 8 + 2 * half + 1];
      A1.x = ldsC[G * 8 + 4 + 2 * half];
      A1.y = ldsC[G * 8 + 5 + 2 * half];

      v8f Cm = {};
      Cm = __builtin_amdgcn_wmma_f32_16x16x4_f32(false, A0, false, B0,
                                                 (short)0, Cm, false, false);
      Cm = __builtin_amdgcn_wmma_f32_16x16x4_f32(false, A1, false, B1,
                                                 (short)0, Cm, false, false);
#pragma unroll
      for (int r = 0; r < 8; ++r) {
        // raw v_sqrt_f32: z feeds a steep sigmoid, 1-ulp sqrt is plenty
        const float z = __builtin_amdgcn_sqrtf(fmaxf(Cm[r], 0.f));
        const float k = sig_of_neg(10.f * (z - 1.f));      // sigmoid(-10(z-1))
        ks += k;
        hs += sig_of_neg(-18.f * (k - 0.1f));              // sigmoid(18(k-0.1))
      }
    }
    // combine the two lane halves (rows 0-7 vs 8-15 of each gaussian tile)
    ks += __shfl_xor(ks, 16);
    hs += __shfl_xor(hs, 16);

    const float ksum = ks + 1e-8f;
    const float H    = (1.f - EMIN_F) * sig_of_neg(10.f * (ksum - 0.5f)) + EMIN_F;
    const float hss  = (1.f - EMIN_F) * hs + (float)NG * EMIN_F;
    const float Hc   = fminf(fmaxf(H, EMIN_F), 1.f);

    acc_comp += mask * (Hc * Hc * Hc * cee);
    acc_H    += mask * H;
    acc_ov   += mask * fmaxf(hss - 1.5f, 0.f);
  }

  block_reduce3(acc_comp, acc_H, acc_ov);
  if (threadIdx.x == 0) {
    part[blockIdx.x * 3 + 0] = acc_comp;
    part[blockIdx.x * 3 + 1] = acc_H;
    part[blockIdx.x * 3 + 2] = acc_ov;
  }
}

// ---------------- kernel 5: final deterministic reduce + objective ----------
__global__ void __launch_bounds__(THREADS) k_final(const float* __restrict__ part,
                                                   const int* __restrict__ gi,
                                                   float* __restrict__ out) {
  float a = 0.f, b = 0.f, c = 0.f;
  for (int i = threadIdx.x; i < MAIN_BLOCKS; i += blockDim.x) {
    a += part[i * 3 + 0];
    b += part[i * 3 + 1];
    c += part[i * 3 + 2];
  }
  block_reduce3(a, b, c);
  if (threadIdx.x == 0) {
    const float Hmean  = b / (float)NEL;
    const float ovmean = c / (float)NEL;
    const float gif    = (float)gi[0];
    const float goal   = 0.3f - 0.1f * fmaxf(0.f, fminf(1.f, (gif - 20.f) / 20.f));
    const float vfloss = fmaxf(Hmean - goal, 0.f);
    out[0] = -a * 1.0f + vfloss * 100.0f + ovmean * 1.0f;
  }
}

extern "C" void kernel_launch(void* const* d_in, const int* in_sizes, int n_in,
                              void* d_out, int out_size, void* d_ws, size_t ws_size,
                              hipStream_t stream) {
  (void)in_sizes; (void)n_in; (void)out_size; (void)ws_size;
  const float* coords = (const float*)d_in[0];
  const float* ce     = (const float*)d_in[1];
  const float* Wscale = (const float*)d_in[2];
  const float* Wshape = (const float*)d_in[3];
  const float* Wrot   = (const float*)d_in[4];
  const float* Woff   = (const float*)d_in[5];
  const int*   gi     = (const int*)d_in[6];
  float* out = (float*)d_out;

  unsigned* hdr = (unsigned*)d_ws;
  float* coef = (float*)((char*)d_ws + WS_COEF_OFF);
  float* part = (float*)((char*)d_ws + WS_PART_OFF);

  k_init<<<1, 32, 0, stream>>>(hdr);
  k_minmax<<<MM_BLOCKS, THREADS, 0, stream>>>(coords, hdr);
  k_coeffs<<<1, NG, 0, stream>>>(Wscale, Wshape, Wrot, Woff, hdr, coef);
  k_main<<<MAIN_BLOCKS, THREADS, 0, stream>>>(coords, ce, coef, part);
  k_final<<<1, THREADS, 0, stream>>>(part, gi, out);
}